// HAvgPool2D_86852828660455
// MI455X (gfx1250) — compile-verified
//
#include <hip/hip_runtime.h>

// ---------------------------------------------------------------------------
// Hyperbolic average pool (Poincare ball, c=1), MI455X / gfx1250, wave32.
//
// Memory-bound problem (~2 flop/byte): input 134MB (fits in 192MB L2),
// output 33.5MB.  Plan:
//   K1: per-input-pixel lambda = 2/max(1 - sum_c x^2, eps), computed with
//       v_wmma_f32_16x16x4_f32 (A = x^2 tile 16 pix x 4 ch, B = ones ->
//       D[m,n] = channel-partial-sum, accumulate 64 chunks).  512KB scratch.
//   K2: per-output-pixel: den/weights from 9 lambda taps (pad tap -> den+=1,
//       weight 0), per-channel weighted 9-tap sum -> v (stored to d_out),
//       sum v^2 -> scale = t/(vn*(1+sqrt(1-t^2)))  [== tanh(0.5*atanh(t))/vn].
//   K3: float4 in-place rescale of d_out by per-pixel scale.
// ---------------------------------------------------------------------------

typedef __attribute__((ext_vector_type(2))) float v2f;
typedef __attribute__((ext_vector_type(4))) float v4f;
typedef __attribute__((ext_vector_type(8))) float v8f;

#define NB   32
#define NC   256
#define NH   64
#define NW   64
#define NOH  32
#define NOW  32
#define HW   (NH * NW)          // 4096
#define OHW  (NOH * NOW)        // 1024
#define NPIX (NB * HW)          // 131072 input pixels
#define NOPIX (NB * OHW)        // 32768 output pixels
#define EPSV 1e-7f
#define TCLIP (1.0f - 1e-5f)

// --------------------------- Kernel 1: lambda ------------------------------
// One wave handles 16 consecutive input pixels; 64 WMMA accumulations over
// the 256-channel K dimension.  A-matrix 16x4 f32 layout (ISA 7.12.2):
//   lanes 0-15 : M=lane,    VGPR0=K0, VGPR1=K1
//   lanes 16-31: M=lane-16, VGPR0=K2, VGPR1=K3
// B = all ones -> D[m,n] = sum_k A[m,k], replicated over all N columns.
__global__ __launch_bounds__(256) void lam_kernel(const float* __restrict__ x,
                                                  float* __restrict__ lam) {
  const int lane  = threadIdx.x & 31;
  const int wave  = (blockIdx.x * blockDim.x + threadIdx.x) >> 5;
  const int m     = lane & 15;
  const int hi    = lane >> 4;            // which half of the wave
  const int pix   = wave * 16 + m;        // waves are 16-aligned, never cross b
  const int b     = pix >> 12;            // / HW
  const int hw    = pix & (HW - 1);
  const float* base = x + (size_t)b * NC * HW + hw;

  v2f ones; ones[0] = 1.0f; ones[1] = 1.0f;
  v8f acc = {};
  const int kofs = hi * 2;
#pragma unroll 4
  for (int ch = 0; ch < NC; ch += 4) {
    float a0 = base[(size_t)(ch + kofs)     * HW];
    float a1 = base[(size_t)(ch + kofs + 1) * HW];
    v2f a; a[0] = a0 * a0; a[1] = a1 * a1;
    // D = A*ones + C : row sums of x^2 chunk, accumulated in C.
    acc = __builtin_amdgcn_wmma_f32_16x16x4_f32(
        false, a, false, ones, (short)0, acc, false, false);
  }

  // acc VGPR r on this lane holds S[r + 8*hi] (replicated over N).
  // We want S[m]: select VGPR (m&7), then pull from the other half-wave if
  // (m>>3) != hi (partner lane computes the same VGPR index).
  const int idx = m & 7;
  float s0 = acc[0], s1 = acc[1], s2 = acc[2], s3 = acc[3];
  float s4 = acc[4], s5 = acc[5], s6 = acc[6], s7 = acc[7];
  float lo4 = (idx & 2) ? ((idx & 1) ? s3 : s2) : ((idx & 1) ? s1 : s0);
  float hi4 = (idx & 2) ? ((idx & 1) ? s7 : s6) : ((idx & 1) ? s5 : s4);
  float mine  = (idx & 4) ? hi4 : lo4;
  float other = __shfl_xor(mine, 16, 32);
  float S = ((m >> 3) == hi) ? mine : other;

  float d = fmaxf(1.0f - S, EPSV);
  lam[pix] = 2.0f / d;
}

// --------------------------- Kernel 2: pool --------------------------------
// Thread per output pixel; lane index = ow so a wave covers one output row
// (input loads are stride-2 within a row -> 2 cachelines / wave / tap, and
// all re-reads of x are L2-resident).
__global__ __launch_bounds__(256) void pool_kernel(const float* __restrict__ x,
                                                   const float* __restrict__ lam,
                                                   float* __restrict__ vout,
                                                   float* __restrict__ scl) {
  const int t  = blockIdx.x * 256 + threadIdx.x;   // < NOPIX
  const int ow = t & 31;
  const int oh = (t >> 5) & 31;
  const int b  = t >> 10;

  const float* lb = lam + b * HW;
  float wgt[9];
  int   off[9];
  float den = 0.0f;
  int q = 0;
#pragma unroll
  for (int kh = 0; kh < 3; ++kh) {
    const int ih = oh * 2 - 1 + kh;
    const bool vh = (ih >= 0) && (ih < NH);
#pragma unroll
    for (int kw = 0; kw < 3; ++kw) {
      const int iw = ow * 2 - 1 + kw;
      const bool ok = vh && (iw >= 0) && (iw < NW);
      const int o = ok ? (ih * NW + iw) : 0;
      const float l = ok ? lb[o] : 0.0f;
      den   += ok ? (l - 1.0f) : 1.0f;   // pad tap: lam=2 -> contributes 1
      wgt[q] = ok ? l : 0.0f;            // pad tap: zero weight on num
      off[q] = o;
      ++q;
    }
  }
  const float inv_den = 1.0f / fmaxf(den, EPSV);

  const float* xb = x + (size_t)b * NC * HW;
  float* vb = vout + (size_t)b * NC * OHW + oh * NOW + ow;
  float sumv2 = 0.0f;
#pragma unroll 2
  for (int c = 0; c < NC; ++c) {
    const float* xc = xb + (size_t)c * HW;
    float num = 0.0f;
#pragma unroll
    for (int k = 0; k < 9; ++k) num = fmaf(wgt[k], xc[off[k]], num);
    const float v = num * inv_den;
    sumv2 = fmaf(v, v, sumv2);
    vb[(size_t)c * OHW] = v;
  }

  const float vn = sqrtf(fmaxf(sumv2, EPSV * EPSV));
  const float tt = fminf(vn, TCLIP);                 // sqrt_c == 1
  // tanh(0.5*atanh(t)) == t / (1 + sqrt(1 - t^2))  (numerically stable)
  scl[t] = tt / (vn * (1.0f + sqrtf(1.0f - tt * tt)));
}

// --------------------------- Kernel 3: rescale -----------------------------
__global__ __launch_bounds__(256) void scale_kernel(float* __restrict__ out,
                                                    const float* __restrict__ scl) {
  const int j = blockIdx.x * 256 + threadIdx.x;      // < out_size/4
  const int i = j * 4;
  // out index: (((b*NC + c)*NOH + oh)*NOW + ow); pixel = b*OHW + (i & (OHW-1))
  const int pixbase = ((i >> 18) << 10) + (i & (OHW - 1));
  v4f o = *(const v4f*)(out + i);
  v4f s = *(const v4f*)(scl + pixbase);
  o *= s;
  *(v4f*)(out + i) = o;
}

// ---------------------------------------------------------------------------
extern "C" void kernel_launch(void* const* d_in, const int* in_sizes, int n_in,
                              void* d_out, int out_size, void* d_ws, size_t ws_size,
                              hipStream_t stream) {
  (void)in_sizes; (void)n_in; (void)out_size; (void)ws_size;
  const float* x = (const float*)d_in[0];
  float* out = (float*)d_out;
  float* lam = (float*)d_ws;               // NPIX floats  (512 KB)
  float* scl = lam + NPIX;                 // NOPIX floats (128 KB)

  // K1: 131072 pixels / 16 per wave = 8192 waves / 8 waves per block.
  lam_kernel<<<NPIX / (16 * 8), 256, 0, stream>>>(x, lam);
  // K2: 32768 output pixels, 256 threads/block.
  pool_kernel<<<NOPIX / 256, 256, 0, stream>>>(x, lam, out, scl);
  // K3: 8388608 outputs / 4 per thread / 256 threads per block.
  scale_kernel<<<(NB * NC * OHW / 4) / 256, 256, 0, stream>>>(out, scl);
}